// GNN_2989297238517
// MI455X (gfx1250) — compile-verified
//
#include <hip/hip_runtime.h>

typedef __attribute__((ext_vector_type(2))) float v2f;
typedef __attribute__((ext_vector_type(8))) float v8f;

#define D 128
#define LDSTRIDE 132   // padded row stride (floats) to avoid LDS bank conflicts

static __device__ __forceinline__ v8f wmma_f32_k4(v2f a, v2f b, v8f c) {
  // D = A(16x4, f32) * B(4x16, f32) + C(16x16, f32)
  return __builtin_amdgcn_wmma_f32_16x16x4_f32(
      /*neg_a=*/false, a, /*neg_b=*/false, b,
      /*c_mod=*/(short)0, c, /*reuse_a=*/false, /*reuse_b=*/false);
}

// ---------------------------------------------------------------- zero ----
__global__ __launch_bounds__(256) void zero_f32(float* __restrict__ p, size_t n) {
  size_t i = (size_t)blockIdx.x * blockDim.x + threadIdx.x;
  size_t stride = (size_t)gridDim.x * blockDim.x;
  for (; i < n; i += stride) p[i] = 0.0f;
}

// --------------------------------------------------------------- degree ---
__global__ __launch_bounds__(256) void degree_kernel(const long long* __restrict__ ei,
                                                     float* __restrict__ cnt, int E) {
  int e = blockIdx.x * blockDim.x + threadIdx.x;
  if (e >= E) return;
  int d = (int)ei[(size_t)E + e];   // dst row
  atomicAdd(&cnt[d], 1.0f);
}

// -------------------------------------------------------------- scatter ---
// One thread per (edge, 4-feature chunk): float4 gather from feat[src], 4
// atomic f32 adds into sum[dst]. x/sum are L2-resident (51.2 MB << 192 MB L2).
__global__ __launch_bounds__(256) void scatter_kernel(const float* __restrict__ feat,
                                                      const long long* __restrict__ ei,
                                                      float* __restrict__ sum, int E) {
  int idx = blockIdx.x * blockDim.x + threadIdx.x;   // E*32 = 51.2M < 2^31
  if (idx >= E * 32) return;
  int e = idx >> 5;
  int c = (idx & 31) << 2;
  int s = (int)ei[e];
  int d = (int)ei[(size_t)E + e];
  const float4 v = *(const float4*)&feat[(size_t)s * D + c];
  float* p = &sum[(size_t)d * D + c];
  atomicAdd(p + 0, v.x);
  atomicAdd(p + 1, v.y);
  atomicAdd(p + 2, v.z);
  atomicAdd(p + 3, v.w);
}

// -------------------------------------------------------------- combine ---
// out[node] = relu( mean[node] @ Wl^T + bl + feat[node] @ Wr^T )
// Block: 128 threads (4 waves), 32 nodes (2 M-tiles of 16). Each wave owns 2
// N-tiles of 16 cols; B fragments are reused across both M-tiles.
__global__ __launch_bounds__(128) void combine_kernel(const float* __restrict__ sum,
                                                      const float* __restrict__ cnt,
                                                      const float* __restrict__ feat,
                                                      const float* __restrict__ Wl,
                                                      const float* __restrict__ bl,
                                                      const float* __restrict__ Wr,
                                                      float* __restrict__ out, int N) {
  __shared__ float sMean[2][16][LDSTRIDE];
  __shared__ float sX[2][16][LDSTRIDE];

  const int m0 = blockIdx.x * 32;

  // ---- stage mean and root activations for 32 nodes into LDS ----
  {
    int t = threadIdx.x;
    int rl = t >> 2;          // local row 0..31
    int seg = t & 3;          // 4 threads per row
    int node = m0 + rl;
    int mt = rl >> 4, mr = rl & 15;
    bool ok = node < N;
    float inv = 1.0f;
    if (ok) inv = 1.0f / fmaxf(cnt[node], 1.0f);
#pragma unroll
    for (int i = 0; i < 8; ++i) {
      int c = seg * 32 + i * 4;
      float4 sv = make_float4(0.f, 0.f, 0.f, 0.f);
      float4 xv = make_float4(0.f, 0.f, 0.f, 0.f);
      if (ok) {
        sv = *(const float4*)&sum[(size_t)node * D + c];
        xv = *(const float4*)&feat[(size_t)node * D + c];
      }
      *(float4*)&sMean[mt][mr][c] =
          make_float4(sv.x * inv, sv.y * inv, sv.z * inv, sv.w * inv);
      *(float4*)&sX[mt][mr][c] = xv;
    }
  }
  __syncthreads();

  const int lane = threadIdx.x & 31;
  const int wave = threadIdx.x >> 5;
  const int n = lane & 15;        // also the A-fragment's M index
  const int g = lane >> 4;        // K half-select per the 16x4 f32 layout

  v8f zc = {0.f, 0.f, 0.f, 0.f, 0.f, 0.f, 0.f, 0.f};
  v8f acc[2][2];
  acc[0][0] = zc; acc[0][1] = zc; acc[1][0] = zc; acc[1][1] = zc;

  const int col0 = (wave * 2 + 0) * 16 + n;   // W row for n-tile 0
  const int col1 = (wave * 2 + 1) * 16 + n;   // W row for n-tile 1

#pragma unroll 4
  for (int k0 = 0; k0 < D; k0 += 4) {
    const int ka = k0 + 2 * g;
    // A fragments: lane (n + 16g) -> act[row=n][ka], act[row=n][ka+1]
    v2f aM0 = *(const v2f*)&sMean[0][n][ka];
    v2f aM1 = *(const v2f*)&sMean[1][n][ka];
    v2f aX0 = *(const v2f*)&sX[0][n][ka];
    v2f aX1 = *(const v2f*)&sX[1][n][ka];
    // B fragments: B[k][col] = W[col][k] (row-major weights, L2 resident)
    v2f bL0 = *(const v2f*)&Wl[(size_t)col0 * D + ka];
    v2f bL1 = *(const v2f*)&Wl[(size_t)col1 * D + ka];
    v2f bR0 = *(const v2f*)&Wr[(size_t)col0 * D + ka];
    v2f bR1 = *(const v2f*)&Wr[(size_t)col1 * D + ka];

    acc[0][0] = wmma_f32_k4(aM0, bL0, acc[0][0]);
    acc[0][0] = wmma_f32_k4(aX0, bR0, acc[0][0]);
    acc[0][1] = wmma_f32_k4(aM0, bL1, acc[0][1]);
    acc[0][1] = wmma_f32_k4(aX0, bR1, acc[0][1]);
    acc[1][0] = wmma_f32_k4(aM1, bL0, acc[1][0]);
    acc[1][0] = wmma_f32_k4(aX1, bR0, acc[1][0]);
    acc[1][1] = wmma_f32_k4(aM1, bL1, acc[1][1]);
    acc[1][1] = wmma_f32_k4(aX1, bR1, acc[1][1]);
  }

  // ---- epilogue: bias + relu + store per C/D layout (VGPR r, lane n+16g
  //      holds element [row = r + 8g][col = n]) ----
  const float bias0 = bl[col0];
  const float bias1 = bl[col1];
#pragma unroll
  for (int mt = 0; mt < 2; ++mt) {
#pragma unroll
    for (int r = 0; r < 8; ++r) {
      int row = m0 + mt * 16 + r + 8 * g;
      if (row < N) {
        float v0 = acc[mt][0][r] + bias0;
        float v1 = acc[mt][1][r] + bias1;
        out[(size_t)row * D + col0] = fmaxf(v0, 0.0f);
        out[(size_t)row * D + col1] = fmaxf(v1, 0.0f);
      }
    }
  }
}

// ---------------------------------------------------------------- launch --
extern "C" void kernel_launch(void* const* d_in, const int* in_sizes, int n_in,
                              void* d_out, int out_size, void* d_ws, size_t ws_size,
                              hipStream_t stream) {
  const float* x        = (const float*)d_in[0];
  const long long* ei   = (const long long*)d_in[1];   // int64 [2, E]
  const float* W1l      = (const float*)d_in[2];
  const float* b1l      = (const float*)d_in[3];
  const float* W1r      = (const float*)d_in[4];
  const float* W2l      = (const float*)d_in[5];
  const float* b2l      = (const float*)d_in[6];
  const float* W2r      = (const float*)d_in[7];
  float* out            = (float*)d_out;

  const int N = in_sizes[0] / D;
  const int E = in_sizes[1] / 2;

  // workspace: [cnt (padded)] [sum N*D] [h N*D]
  const size_t cntPad = (size_t)((N + 255) & ~255);
  float* cnt = (float*)d_ws;
  float* sum = cnt + cntPad;
  float* h   = sum + (size_t)N * D;

  const int zb = 2048;
  const int scatterBlocks = (E * 32 + 255) / 256;
  const int combineBlocks = (N + 31) / 32;

  // degrees (shared by both layers) + clear sums
  zero_f32<<<zb, 256, 0, stream>>>(cnt, cntPad + (size_t)N * D);
  degree_kernel<<<(E + 255) / 256, 256, 0, stream>>>(ei, cnt, E);

  // ---- layer 1 ----
  scatter_kernel<<<scatterBlocks, 256, 0, stream>>>(x, ei, sum, E);
  combine_kernel<<<combineBlocks, 128, 0, stream>>>(sum, cnt, x, W1l, b1l, W1r, h, N);

  // ---- layer 2 ----
  zero_f32<<<zb, 256, 0, stream>>>(sum, (size_t)N * D);
  scatter_kernel<<<scatterBlocks, 256, 0, stream>>>(h, ei, sum, E);
  combine_kernel<<<combineBlocks, 128, 0, stream>>>(sum, cnt, h, W2l, b2l, W2r, out, N);
}